// Sinusoidal_75960791597375
// MI455X (gfx1250) — compile-verified
//
#include <hip/hip_runtime.h>
#include <math.h>

typedef __attribute__((ext_vector_type(2))) float v2f;
typedef __attribute__((ext_vector_type(8))) float v8f;

#define NB   4
#define NT   250
#define NN   100
#define NDEP 64
#define HOP  256
#define NSAMP 64000

#define PI_F        3.14159265358979323846f
#define TWO_PI_D    6.283185307179586
#define OMEGA_SCALE 3.9269908169872414e-4f   // 2*pi/16000
#define LOG2_400    8.6438561897747244f      // log2(HZ_MAX/HZ_MIN)
#define LN10        2.3025850929940457f

// ---------------------------------------------------------------------------
// Kernel 1: get_controls. One thread per (b,t,n) softmax-of-64 + exp_sigmoid.
// Emits per-frame omega (rad/sample) and nyquist-masked amplitude.
// ---------------------------------------------------------------------------
__global__ void __launch_bounds__(256)
controls_kernel(const float* __restrict__ amp_logits,   // [NB,NT,NN]
                const float* __restrict__ freq_logits,  // [NB,NT,NN*NDEP]
                float* __restrict__ omega,              // [NB,NT,NN]
                float* __restrict__ amp) {              // [NB,NT,NN]
    int idx = blockIdx.x * blockDim.x + threadIdx.x;
    if (idx >= NB * NT * NN) return;

    const float4* lg = (const float4*)(freq_logits + (size_t)idx * NDEP);

    float x[NDEP];
    float m = -INFINITY;
#pragma unroll
    for (int i = 0; i < NDEP / 4; ++i) {
        float4 q = lg[i];
        x[4*i+0] = q.x; x[4*i+1] = q.y; x[4*i+2] = q.z; x[4*i+3] = q.w;
        m = fmaxf(m, fmaxf(fmaxf(q.x, q.y), fmaxf(q.z, q.w)));
    }
    float s = 0.0f, sw = 0.0f;
#pragma unroll
    for (int d = 0; d < NDEP; ++d) {
        float e = __expf(x[d] - m);
        s  += e;
        sw += e * (float)d;
    }
    // softmax expectation over linspace(0,1,64); bin(d) = d/63
    float f_unit = sw / (s * 63.0f);
    // unit_to_hz: 20 * 2^(u * log2(8000/20))
    float hz = 20.0f * exp2f(f_unit * LOG2_400);

    // exp_sigmoid(amp) = 2 * sigmoid(a)^ln(10) + 1e-7
    float a_in = amp_logits[idx];
    float sig  = 1.0f / (1.0f + __expf(-a_in));
    float a    = 2.0f * __powf(sig, LN10) + 1e-7f;
    if (!(hz < 8000.0f)) a = 0.0f;   // remove_above_nyquist

    omega[idx] = hz * OMEGA_SCALE;
    amp[idx]   = a;
}

// ---------------------------------------------------------------------------
// Kernel 2: closed-form replacement for the 64000-long cumsum.
// omega is piecewise-linear per 256-sample segment, so the segment sum is
//   S_k = 256*w0 + 127.5*(w1 - w0)
// Scan 250 frames per (b,n) channel in double, store P_k mod 2pi as f32.
// ---------------------------------------------------------------------------
__global__ void phase_scan_kernel(const float* __restrict__ omega,   // [NB,NT,NN]
                                  float* __restrict__ pbase) {       // [NB,NT,NN]
    int idx = blockIdx.x * blockDim.x + threadIdx.x;   // over NB*NN channels
    if (idx >= NB * NN) return;
    int b = idx / NN, n = idx % NN;
    const float* om = omega + (size_t)b * NT * NN + n;
    float*       pb = pbase + (size_t)b * NT * NN + n;

    double acc = 0.0;
    for (int k = 0; k < NT; ++k) {
        pb[k * NN] = (float)fmod(acc, TWO_PI_D);
        float w0 = om[k * NN];
        float w1 = om[min(k + 1, NT - 1) * NN];
        acc += 256.0 * (double)w0 + 127.5 * (double)(w1 - w0);
    }
}

// ---------------------------------------------------------------------------
// Kernel 3: synthesis. One block per (b, frame); 16 waves x 16 samples = 256
// samples/segment. Per-wave reduction over the 100 sinusoids is done with
// V_WMMA_F32_16X16X4_F32: A[16 samp x 4 chan] = amp*sin terms, B has a ones
// column, C accumulates across 25 channel chunks. Column 0 of D = output.
// ---------------------------------------------------------------------------
__global__ void __launch_bounds__(512)
synth_kernel(const float* __restrict__ omega,
             const float* __restrict__ amp,
             const float* __restrict__ pbase,
             float* __restrict__ out) {                 // [NB, NSAMP]
    __shared__ float sw0[NN], sdw[NN], spb[NN], sa0[NN], sda[NN];

    int b  = blockIdx.x / NT;
    int k  = blockIdx.x % NT;
    int kp = min(k + 1, NT - 1);
    const size_t base0 = ((size_t)b * NT + k)  * NN;
    const size_t base1 = ((size_t)b * NT + kp) * NN;

    for (int n = threadIdx.x; n < NN; n += blockDim.x) {
        float w0 = omega[base0 + n];
        float w1 = omega[base1 + n];
        sw0[n] = w0;
        sdw[n] = w1 - w0;
        spb[n] = pbase[base0 + n];
        float a0 = amp[base0 + n];
        float a1 = amp[base1 + n];
        sa0[n] = a0;
        sda[n] = a1 - a0;
    }
    __syncthreads();

    int lane = threadIdx.x & 31;
    int wid  = threadIdx.x >> 5;        // 0..15 -> 16-sample tile
    int m    = lane & 15;               // matrix row M = sample in tile
    int hi   = lane >> 4;               // 0: K=0,1   1: K=2,3
    int r    = wid * 16 + m;            // sample within segment, 0..255

    // phase(r) = P + (r+1)*w0 + dw * r(r+1)/512 ; amp(r) = a0 + da*hann(r)
    float c1 = (float)(r + 1);
    float c2 = (float)(r * (r + 1)) * (1.0f / 512.0f);
    float hw = 0.5f - 0.5f * __cosf(PI_F * (float)r * (1.0f / 256.0f));

    // B matrix: ones in column N==0, zero elsewhere (layout-robust: every
    // K-row slot for N==0 lives on a lane with (lane&15)==0).
    float be = (m == 0) ? 1.0f : 0.0f;
    v2f bvec = {be, be};

    v8f acc = {};
#pragma unroll
    for (int c = 0; c < NN / 4; ++c) {
        int n0 = 4 * c + 2 * hi;
        int n1 = n0 + 1;
        v2f avec;
        {
            float a  = sa0[n0] + sda[n0] * hw;
            float ph = spb[n0] + c1 * sw0[n0] + c2 * sdw[n0];
            avec.x = a * __sinf(ph);
        }
        {
            float a  = sa0[n1] + sda[n1] * hw;
            float ph = spb[n1] + c1 * sw0[n1] + c2 * sdw[n1];
            avec.y = a * __sinf(ph);
        }
        acc = __builtin_amdgcn_wmma_f32_16x16x4_f32(
            /*neg_a=*/false, avec, /*neg_b=*/false, bvec,
            /*c_mod=*/(short)0, acc, /*reuse_a=*/false, /*reuse_b=*/false);
    }

    // D column 0: rows 0..7 in lane 0 (VGPR0..7), rows 8..15 in lane 16.
    if (m == 0) {
        float* o = out + (size_t)b * NSAMP + (size_t)k * HOP + wid * 16 + hi * 8;
#pragma unroll
        for (int j = 0; j < 8; ++j) o[j] = acc[j];
    }
}

// ---------------------------------------------------------------------------
extern "C" void kernel_launch(void* const* d_in, const int* in_sizes, int n_in,
                              void* d_out, int out_size, void* d_ws, size_t ws_size,
                              hipStream_t stream) {
    const float* amp_logits  = (const float*)d_in[0];  // [4,250,100]
    const float* freq_logits = (const float*)d_in[1];  // [4,250,6400]
    float* ws    = (float*)d_ws;
    float* omega = ws;                         // 100000 f32
    float* amp   = ws + (size_t)NB * NT * NN;  // 100000 f32
    float* pbase = ws + 2ull * NB * NT * NN;   // 100000 f32
    float* out   = (float*)d_out;              // [4,64000]

    controls_kernel<<<(NB * NT * NN + 255) / 256, 256, 0, stream>>>(
        amp_logits, freq_logits, omega, amp);
    phase_scan_kernel<<<(NB * NN + 127) / 128, 128, 0, stream>>>(omega, pbase);
    synth_kernel<<<NB * NT, 512, 0, stream>>>(omega, amp, pbase, out);
}